// LIFNeuron_67250597921098
// MI455X (gfx1250) — compile-verified
//
#include <hip/hip_runtime.h>
#include <cmath>

// LIF forward scan: x_seq (T=64, B=32, F=8192) f32 -> (spike_seq, mem_seq).
// Memory-bound streaming kernel: depth-14 async global->LDS pipeline per wave
// (gfx1250 GLOBAL_LOAD_ASYNC_TO_LDS_B128 + s_wait_asynccnt), b128 NT stores.

typedef float v4f __attribute__((ext_vector_type(4)));

#define T_STEPS 64
#define BLOCK   256
#define DEPTH   14
#define SLOTS   (DEPTH + 1)   // ring slots; SLOTS = DEPTH+1 => write slot != read slot

#define STR1(x) #x
#define STR(x)  STR1(x)

__device__ __forceinline__ void lif_step(v4f& mem, const v4f& xv, float beta, v4f& s) {
#pragma unroll
  for (int c = 0; c < 4; ++c) {
    float m = fmaf(mem[c], beta, xv[c]);     // mem*beta + x
    float fire = (m >= 1.0f) ? 1.0f : 0.0f;  // Heaviside(mem - V_th >= 0)
    s[c] = fire;
    mem[c] = (m >= 1.0f) ? 0.0f : m;         // hard reset to V_RESET=0
  }
}

__global__ __launch_bounds__(BLOCK) void lif_fwd_async(const float* __restrict__ x,
                                                       float* __restrict__ spk,
                                                       float* __restrict__ memo,
                                                       float beta, int n4) {
  __shared__ v4f lbuf[SLOTS][BLOCK];         // 15 * 256 * 16B = 60 KiB

  const int tid = threadIdx.x;
  const long long gid = (long long)blockIdx.x * BLOCK + tid;   // float4 index in one timestep
  if (gid >= n4) return;

  const v4f* xg = reinterpret_cast<const v4f*>(x) + gid;
  v4f* sg = reinterpret_cast<v4f*>(spk) + gid;
  v4f* mg = reinterpret_cast<v4f*>(memo) + gid;

  // Low 32 bits of a generic LDS pointer are the LDS byte offset (aperture in hi bits).
  const unsigned lds_base = (unsigned)(unsigned long long)(&lbuf[0][tid]);
  const unsigned slot_stride = (unsigned)(BLOCK * sizeof(v4f));  // 4096 B

  // ---- prologue: fill pipeline with timesteps 0..DEPTH-1 ----
#pragma unroll
  for (int i = 0; i < DEPTH; ++i) {
    const v4f* src = xg + (long long)i * n4;
    unsigned laddr = lds_base + (unsigned)i * slot_stride;
    asm volatile("global_load_async_to_lds_b128 %0, %1, off"
                 :: "v"(laddr), "v"(src) : "memory");
  }

  v4f mem = {0.0f, 0.0f, 0.0f, 0.0f};        // V_REST = 0
  int slot_rd = 0;
  int slot_wr = DEPTH;

  // ---- main loop: refill DEPTH ahead, consume oldest ----
  for (int t = 0; t <= T_STEPS - 1 - DEPTH; ++t) {
    {
      const v4f* src = xg + (long long)(t + DEPTH) * n4;
      unsigned laddr = lds_base + (unsigned)slot_wr * slot_stride;
      asm volatile("global_load_async_to_lds_b128 %0, %1, off"
                   :: "v"(laddr), "v"(src) : "memory");
      if (++slot_wr == SLOTS) slot_wr = 0;
    }
    // Outstanding after issue <= DEPTH+1; loads complete in order, so
    // ASYNCcnt <= DEPTH guarantees timestep t has landed in LDS.
    asm volatile("s_wait_asynccnt " STR(DEPTH) ::: "memory");

    v4f xv = lbuf[slot_rd][tid];             // ds_load_b128
    if (++slot_rd == SLOTS) slot_rd = 0;

    v4f s;
    lif_step(mem, xv, beta, s);
    __builtin_nontemporal_store(s,   sg + (long long)t * n4);
    __builtin_nontemporal_store(mem, mg + (long long)t * n4);
  }

  // ---- drain: all 64 loads issued; wait for everything, consume the rest ----
  asm volatile("s_wait_asynccnt 0x0" ::: "memory");
#pragma unroll
  for (int k = 0; k < DEPTH; ++k) {
    const int t = T_STEPS - DEPTH + k;
    v4f xv = lbuf[slot_rd][tid];
    if (++slot_rd == SLOTS) slot_rd = 0;

    v4f s;
    lif_step(mem, xv, beta, s);
    __builtin_nontemporal_store(s,   sg + (long long)t * n4);
    __builtin_nontemporal_store(mem, mg + (long long)t * n4);
  }
}

extern "C" void kernel_launch(void* const* d_in, const int* in_sizes, int n_in,
                              void* d_out, int out_size, void* d_ws, size_t ws_size,
                              hipStream_t stream) {
  (void)n_in; (void)d_ws; (void)ws_size; (void)out_size;

  const float* x = (const float*)d_in[0];
  const int total = in_sizes[0];             // T*B*F = 16,777,216
  const int per_t = total / T_STEPS;         // B*F  = 262,144
  const int n4 = per_t / 4;                  // float4 per timestep = 65,536

  float* spk  = (float*)d_out;               // spike_seq, T*B*F floats
  float* memo = (float*)d_out + (size_t)total; // mem_seq,  T*B*F floats

  // beta = exp(-dt / (|tau| + 1e-6)) with dt=1, tau=2 (reference constants)
  const float beta = expf(-1.0f / (fabsf(2.0f) + 1e-6f));

  const int grid = (n4 + BLOCK - 1) / BLOCK; // 256 blocks x 256 threads (8 waves)
  lif_fwd_async<<<grid, BLOCK, 0, stream>>>(x, spk, memo, beta, n4);
}